// JConfigurationErrorBlock_70205535421260
// MI455X (gfx1250) — compile-verified
//
#include <hip/hip_runtime.h>

typedef __attribute__((ext_vector_type(2))) float v2f;
typedef __attribute__((ext_vector_type(8))) float v8f;
typedef __attribute__((ext_vector_type(4))) int   v4i;

// Problem constants (from reference): B=32, N=200, T=20000, S=3N=600
constexpr int  BQ   = 32;
constexpr int  TT   = 20000;
constexpr int  SDIM = 600;
constexpr int  TRIPS_PER_WAVE  = 4;     // divides TT -> a wave never crosses z
constexpr int  WAVES_PER_BLOCK = 8;     // 256 threads = 8 wave32

__device__ __forceinline__ float bperm(int byteIdx, float v) {
    return __builtin_bit_cast(float,
        __builtin_amdgcn_ds_bpermute(byteIdx, __builtin_bit_cast(int, v)));
}

// One wave handles one (z,t) triplet per unrolled iteration with two
// V_WMMA_F32_16X16X4_F32 ops:
//   stage1: D1 = [Y0;Y1](6x3) x [Z0|Z1](3x6)   -> W_bc = Yb*Zc at block (3b,3c)
//   stage2: D2 = [X0;X1](6x3) x B2(3x12) + C   -> P_abc - I  (delta folded in C)
// Sign variant: M1[p][q] = s[p]*s[q]*M[p][q], s = [1,1,-1].
// The wave id is readfirstlane'd so every per-triplet base address is provably
// wave-uniform: triplet indices come via SMEM, H loads and out stores use
// scalar-base + 32-bit lane-offset (saddr) addressing, and the unrolled body
// carries no vector 64-bit address arithmetic.
__global__ __launch_bounds__(256)
void jconfig_err_kernel(const float* __restrict__ H,
                        const int*   __restrict__ trip,
                        float*       __restrict__ out)
{
    const int lane = threadIdx.x & 31;
    // wave index within block is wave-uniform: make it provably scalar
    const int wib  = __builtin_amdgcn_readfirstlane((int)(threadIdx.x >> 5));
    const int gw   = blockIdx.x * WAVES_PER_BLOCK + wib;

    const int half = lane >> 4;     // 0: K/rows {0,1}; 1: K/rows {2,3}
    const int m    = lane & 15;     // A-row / B-col index

    // ---- [M0;M1]/[Z0|Z1] sign-stack factors as one-hot coefficients ----
    const bool  act6 = (m < 6);
    const int   pr   = m % 3;
    const float sPr  = (pr == 2) ? -1.f : 1.f;             // s[pr]
    const float fH0  = (m >= 3) ?  sPr : 1.f;              // q/row in {0,1}
    const float fH1  = (m >= 3) ? -sPr : 1.f;              // q/row == 2
    const float facX = act6 ? (half ? fH1 : fH0) : 0.f;
    float facA0 = half ? 0.f : facX;                       // elem.x <- v0
    float facA1 = half ? facX : 0.f;                       // elem.x <- v1
    float facYc = (act6 && !half) ? fH0 : 0.f;             // elem.y <- v1

    // loop-invariant per-lane load offsets (float elements)
    int laneA = pr * SDIM + half;                          // row loads (X, Y)
    int laneB = half * SDIM + pr;                          // col loads (Z)

    // ---- stage-2 B rearrangement / norm constants (cols 0..11 live) ----
    const int  g2 = m / 3, s2 = m % 3;                     // g2 = 2b+c
    const bool bb = ((g2 >> 1) & 1) != 0;
    const bool live12 = (m < 12);
    // b2.x = u0*c00 + u3*c01 + u2*c02 + u5*c03 ; b2.y = u1*c00 + u4*c01
    float c00 = (live12 && !half && !bb) ? 1.f : 0.f;
    float c01 = (live12 && !half &&  bb) ? 1.f : 0.f;
    float c02 = (live12 &&  half && !bb) ? 1.f : 0.f;
    float c03 = (live12 &&  half &&  bb) ? 1.f : 0.f;

    int srcB  = (3 * (g2 & 1) + s2) * 4;                   // bpermute byte idx
    int rot1B = (3 * g2 + ((s2 + 1) % 3)) * 4;
    int rot2B = (3 * g2 + ((s2 + 2) % 3)) * 4;

    // stage-2 WMMA C operand: -I replicated per 3x3 block (rows = VGPR axis)
    float cm0 = (s2 == 0) ? -1.f : 0.f;
    float cm1 = (s2 == 1) ? -1.f : 0.f;
    float cm2 = (s2 == 2) ? -1.f : 0.f;

    const bool doStore = (lane < 12) && (s2 == 0);

    // Pin all loop invariants into VGPRs: empty asm is free but makes the
    // values opaque, so the unrolled loop cannot rematerialize them.
    asm volatile("" : "+v"(facA0), "+v"(facA1), "+v"(facYc),
                      "+v"(c00), "+v"(c01), "+v"(c02), "+v"(c03),
                      "+v"(cm0), "+v"(cm1), "+v"(cm2),
                      "+v"(srcB), "+v"(rot1B), "+v"(rot2B),
                      "+v"(laneA), "+v"(laneB));

    const v8f c2 = {cm0, cm1, cm2, cm0, cm1, cm2, 0.f, 0.f};

    // wave -> (z, t0): TT % TRIPS_PER_WAVE == 0, so z is fixed per wave
    const int z  = gw / (TT / TRIPS_PER_WAVE);
    const int t0 = (gw % (TT / TRIPS_PER_WAVE)) * TRIPS_PER_WAVE;
    const float* Hz = H + (size_t)z * SDIM * SDIM;         // scalar base
    float* outZ = out + ((size_t)z * TT + t0) * 8;         // scalar base

    // all 12 triplet indices for this wave: uniform address -> SMEM s_load
    const v4i q0 = *(const v4i*)(trip + 3 * t0);
    const v4i q1 = *(const v4i*)(trip + 3 * t0 + 4);
    const v4i q2 = *(const v4i*)(trip + 3 * t0 + 8);
    const int idx[12] = {q0.x, q0.y, q0.z, q0.w, q1.x, q1.y,
                         q1.z, q1.w, q2.x, q2.y, q2.z, q2.w};

    #pragma unroll
    for (int it = 0; it < TRIPS_PER_WAVE; ++it) {
        const int i3 = 3 * idx[3 * it + 0];
        const int j3 = 3 * idx[3 * it + 1];
        const int k3 = 3 * idx[3 * it + 2];

        // scalar block bases (SALU); per-lane part is the pinned voffset
        const float* pY = Hz + j3 * SDIM + k3;   // Y = Hb[j,k]
        const float* pZ = Hz + k3 * SDIM + i3;   // Z = Hb[k,i]
        const float* pX = Hz + i3 * SDIM + j3;   // X = Hb[i,j]

        const float y0 = pY[laneA], y1 = pY[laneA + 1];     // saddr + b64
        const float z0 = pZ[laneB], z1 = pZ[laneB + SDIM];  // saddr + 2 x b32
        const float x0 = pX[laneA], x1 = pX[laneA + 1];     // saddr + b64

        v2f a1, b1, a2;
        a1.x = fmaf(y0, facA0, y1 * facA1);  a1.y = y1 * facYc;
        b1.x = fmaf(z0, facA0, z1 * facA1);  b1.y = z1 * facYc;
        a2.x = fmaf(x0, facA0, x1 * facA1);  a2.y = x1 * facYc;

        const v8f zero8 = {};
        // D1[3b+r, 3c+s] = (Yb*Zc)[r][s]
        const v8f d1 = __builtin_amdgcn_wmma_f32_16x16x4_f32(
            false, a1, false, b1, (short)0, zero8, false, false);

        // B2[r, 3*(2b+c)+s] = D1[3b+r, 3c+s] via one-hot blend of bpermutes
        const float u0 = bperm(srcB, d1[0]);
        const float u1 = bperm(srcB, d1[1]);
        const float u2 = bperm(srcB, d1[2]);
        const float u3 = bperm(srcB, d1[3]);
        const float u4 = bperm(srcB, d1[4]);
        const float u5 = bperm(srcB, d1[5]);
        v2f b2;
        b2.x = fmaf(u0, c00, fmaf(u3, c01, fmaf(u2, c02, u5 * c03)));
        b2.y = fmaf(u1, c00, u4 * c01);

        // D2 = [X0;X1] x B2 - I   (delta in C operand)
        const v8f d2 = __builtin_amdgcn_wmma_f32_16x16x4_f32(
            false, a2, false, b2, (short)0, c2, false, false);

        // Frobenius norms: per-lane partial over p (VGPR axis), then
        // rotation-sum over the 3 column lanes of each group.
        const float pa0 = fmaf(d2[0], d2[0], fmaf(d2[1], d2[1], d2[2] * d2[2]));
        const float pa1 = fmaf(d2[3], d2[3], fmaf(d2[4], d2[4], d2[5] * d2[5]));
        const float n0 = pa0 + bperm(rot1B, pa0) + bperm(rot2B, pa0);
        const float n1 = pa1 + bperm(rot1B, pa1) + bperm(rot2B, pa1);

        if (doStore) {
            float* ob = outZ + it * 8;          // scalar base, lane voffset g2
            ob[g2]     = __builtin_amdgcn_sqrtf(n0);   // a = 0
            ob[4 + g2] = __builtin_amdgcn_sqrtf(n1);   // a = 1
        }
    }
}

extern "C" void kernel_launch(void* const* d_in, const int* in_sizes, int n_in,
                              void* d_out, int out_size, void* d_ws, size_t ws_size,
                              hipStream_t stream)
{
    (void)in_sizes; (void)n_in; (void)out_size; (void)d_ws; (void)ws_size;
    const float* H    = (const float*)d_in[0];   // (32, 600, 600) f32
    const int*   trip = (const int*)d_in[1];     // (20000, 3) i32
    float*       out  = (float*)d_out;           // (32, 20000, 8) f32

    const long totalWaves = ((long)BQ * TT) / TRIPS_PER_WAVE;          // 160000
    const int  blocks     = (int)(totalWaves / WAVES_PER_BLOCK);       // 20000

    jconfig_err_kernel<<<blocks, WAVES_PER_BLOCK * 32, 0, stream>>>(H, trip, out);
}